// TtMixtralAttention_73521250173386
// MI455X (gfx1250) — compile-verified
//
#include <hip/hip_runtime.h>
#include <hip/hip_bf16.h>
#include <math.h>

// ---------------------------------------------------------------------------
// Mixtral attention decode step, fp32, CDNA5 (gfx1250) WMMA version.
// B=32, H=4096, NH=32, NKV=8, D=128, W=2048, G=4. Bandwidth-bound (~680MB).
// Uses V_WMMA_F32_16X16X4_F32 for all GEMMs (native fp32 matrix op).
// ---------------------------------------------------------------------------

typedef __attribute__((ext_vector_type(2))) float v2f;
typedef __attribute__((ext_vector_type(8))) float v8f;

#define BATCH  32
#define HDIM   4096
#define NH     32
#define NKV    8
#define HEADD  128
#define WCTX   2048
#define GRP    4
#define NQKV   (NH*HEADD + 2*NKV*HEADD)   // 6144
#define SCALE  0.08838834764831845f       // 1/sqrt(128)

static __device__ __forceinline__ v8f wmma_f32_k4(v2f a, v2f b, v8f c) {
    // D = A(16x4 f32) * B(4x16 f32) + C(16x16 f32)
    return __builtin_amdgcn_wmma_f32_16x16x4_f32(
        false, a, false, b, (short)0, c, false, false);
}

// ---------------------------------------------------------------------------
// Kernel 1: fused QKV projection + RoPE.  C = x(32x4096) @ [wq|wk|wv]^T.
// One wave per 16x16 tile; K marched in steps of 4 via fp32 WMMA.
// q -> q_buf[32][4096] (rope'd), k -> k_new[32][1024] (rope'd), v -> v_new.
// ---------------------------------------------------------------------------
__global__ __launch_bounds__(256)
void qkv_rope_kernel(const float* __restrict__ x,
                     const float* __restrict__ wq,
                     const float* __restrict__ wk,
                     const float* __restrict__ wv,
                     const float* __restrict__ rot,
                     float* __restrict__ q_buf,
                     float* __restrict__ k_new,
                     float* __restrict__ v_new)
{
    const int wave = threadIdx.x >> 5;
    const int lane = threadIdx.x & 31;
    const int tile = blockIdx.x * 8 + wave;          // 0..767
    const int mtile = tile & 1;                      // 2 M tiles (B=32)
    const int ntile = tile >> 1;                     // 384 N tiles (N=6144)
    const int mbase = mtile * 16;
    const int nbase = ntile * 16;
    const int n16   = lane & 15;
    const int c2    = (lane & 16) ? 2 : 0;           // K sub-offset per half-wave
    const int col   = nbase + n16;                   // global output column

    // Which weight matrix does this tile belong to? (tile never straddles)
    const float* wrow;
    int mode;                                        // 0=q 1=k 2=v
    if (col < NH*HEADD)                { wrow = wq + (size_t)col * HDIM;                mode = 0; }
    else if (col < NH*HEADD+NKV*HEADD) { wrow = wk + (size_t)(col - NH*HEADD) * HDIM;   mode = 1; }
    else                               { wrow = wv + (size_t)(col - NH*HEADD - NKV*HEADD) * HDIM; mode = 2; }

    const float* arow = x + (size_t)(mbase + n16) * HDIM + c2;  // A: x rows
    const float* brow = wrow + c2;                              // B: W^T cols

    v8f acc = {};
    #pragma unroll 8
    for (int k = 0; k < HDIM; k += 4) {
        v2f a = *(const v2f*)(arow + k);   // A[m=n16, k0+c2 .. +1]
        v2f b = *(const v2f*)(brow + k);   // B[k0+c2 .. +1, n=n16]
        acc = wmma_f32_k4(a, b, acc);
    }

    // RoPE on q and k tiles: pair (2j,2j+1) within head dim lives in adjacent
    // lanes; exchange with shfl_xor(1) and apply 2x2 rotation.
    if (mode != 2) {
        const int d = col & (HEADD - 1);
        const int i = d & ~1;
        const float cth = rot[i * HEADD + i];
        const float sth = rot[i * HEADD + i + 1];
        #pragma unroll
        for (int r = 0; r < 8; ++r) {
            float own   = acc[r];
            float other = __shfl_xor(own, 1, 32);
            acc[r] = (d & 1) ? (sth * other + cth * own)
                             : (cth * own - sth * other);
        }
    }

    // C/D layout: VGPR r -> row (mbase + r) lanes 0-15, (mbase + r + 8) lanes 16-31
    const int rowoff = (lane & 16) ? 8 : 0;
    #pragma unroll
    for (int r = 0; r < 8; ++r) {
        const int row = mbase + r + rowoff;
        if (mode == 0)       q_buf[(size_t)row * (NH*HEADD)  + col]                      = acc[r];
        else if (mode == 1)  k_new[(size_t)row * (NKV*HEADD) + (col - NH*HEADD)]         = acc[r];
        else                 v_new[(size_t)row * (NKV*HEADD) + (col - NH*HEADD - NKV*HEADD)] = acc[r];
    }
}

// ---------------------------------------------------------------------------
// Kernel 2: attention for one (kv, b). 8 waves split P=2048 (256 pos each).
//  pass1: scores  C[m=g(pad16), n=p16] = q_pad(16x128) x keys^T, +mask, *scale
//  pass2: block softmax over P (max / exp / sum) in LDS
//  pass3: attn    C[m=g(pad16), n=d16] = probs x vals, cross-wave LDS reduce
// Row current_pos is sourced from k_new / v_new (inputs are immutable).
// ---------------------------------------------------------------------------
#define QS 132          // padded q row stride (bank-conflict avoidance)
#define SS 2052         // padded score row stride

__global__ __launch_bounds__(256)
void attn_kernel(const float* __restrict__ cache_k,
                 const float* __restrict__ cache_v,
                 const float* __restrict__ attn_mask,
                 const int*   __restrict__ start_pos_p,
                 const int*   __restrict__ current_pos_p,
                 const float* __restrict__ q_buf,
                 const float* __restrict__ k_new,
                 const float* __restrict__ v_new,
                 float* __restrict__ attn_buf)
{
    __shared__ float q_pad[16 * QS];          //  ~8.4 KB (rows 4..15 zero)
    __shared__ float s_lds[4 * SS];           // ~32.8 KB scores -> probs
    __shared__ float attn_red[8][4][HEADD];   //  16.0 KB per-wave partials
    __shared__ float red[4][64];              //   1.0 KB reduction scratch
    __shared__ float m_g[4], inv_l[4];

    const int kv   = blockIdx.x;
    const int b    = blockIdx.y;
    const int tid  = threadIdx.x;
    const int wave = tid >> 5;
    const int lane = tid & 31;
    const int n16  = lane & 15;
    const int c2   = (lane & 16) ? 2 : 0;

    const int cur = current_pos_p[0];
    const int sp  = start_pos_p[0];
    int P = ((sp + 1 + 31) >> 5) << 5;
    if (P > WCTX) P = WCTX;

    // Stage q (4 heads of this kv group) into LDS, zero-padded to 16 rows.
    for (int i = tid; i < 16 * HEADD; i += 256) {
        const int g = i >> 7, d = i & (HEADD - 1);
        q_pad[g * QS + d] = (g < GRP)
            ? q_buf[(size_t)b * (NH*HEADD) + (kv * GRP + g) * HEADD + d] : 0.f;
    }
    __syncthreads();

    const size_t kbase = ((size_t)kv * BATCH + b) * WCTX * HEADD;
    const float* knew_row = k_new + (size_t)b * (NKV*HEADD) + kv * HEADD;
    const float* vnew_row = v_new + (size_t)b * (NKV*HEADD) + kv * HEADD;

    // ---- pass 1: scores --------------------------------------------------
    for (int t = 0; t < 16; ++t) {
        const int pbase = wave * 256 + t * 16;
        const int pl    = pbase + n16;                 // this lane's position
        const float* krow = (pl == cur) ? knew_row
                                        : (cache_k + kbase + (size_t)pl * HEADD);
        if (t + 1 < 16)   // speculative prefetch of next tile's key row
            __builtin_prefetch(cache_k + kbase + (size_t)(pl + 16) * HEADD, 0, 0);
        const float* kc = krow + c2;

        v8f acc = {};
        #pragma unroll
        for (int k0 = 0; k0 < HEADD; k0 += 4) {
            v2f a = *(const v2f*)(&q_pad[n16 * QS + k0 + c2]); // A[m=g,k]
            v2f kb = *(const v2f*)(kc + k0);                   // B[k,n=p]
            acc = wmma_f32_k4(a, kb, acc);
        }
        if (lane < 16) {
            const int p = pbase + lane;
            #pragma unroll
            for (int r = 0; r < GRP; ++r) {
                float sv = acc[r] * SCALE
                         + attn_mask[((size_t)b * NH + (kv * GRP + r)) * WCTX + p];
                if (p >= P) sv = -1e30f;
                s_lds[r * SS + p] = sv;
            }
        }
    }
    __syncthreads();

    // ---- pass 2: softmax -------------------------------------------------
    {
        const int g = tid >> 6, l64 = tid & 63;
        float m = -3.0e38f;
        for (int p = l64; p < WCTX; p += 64) m = fmaxf(m, s_lds[g * SS + p]);
        red[g][l64] = m;
        __syncthreads();
        if (tid < 4) {
            float mm = -3.0e38f;
            for (int j = 0; j < 64; ++j) mm = fmaxf(mm, red[tid][j]);
            m_g[tid] = mm;
        }
        __syncthreads();
        const float mm = m_g[g];
        float s = 0.f;
        for (int p = l64; p < WCTX; p += 64) {
            float e = __expf(s_lds[g * SS + p] - mm);
            s_lds[g * SS + p] = e;
            s += e;
        }
        red[g][l64] = s;
        __syncthreads();
        if (tid < 4) {
            float tot = 0.f;
            for (int j = 0; j < 64; ++j) tot += red[tid][j];
            inv_l[tid] = 1.0f / tot;
        }
        __syncthreads();
    }

    // ---- pass 3: attn = probs @ vals ------------------------------------
    const int kslice = wave * 256;
    for (int nt = 0; nt < 8; ++nt) {
        const int dbase = nt * 16;
        v8f acc = {};
        #pragma unroll 4
        for (int k = kslice; k < kslice + 256; k += 4) {
            const int p0 = k + c2, p1 = k + c2 + 1;
            v2f a;  // A[m=g, k=p] ; rows g>=4 are zero
            {
                float a0 = s_lds[(n16 & 3) * SS + p0];
                float a1 = s_lds[(n16 & 3) * SS + p1];
                a.x = (n16 < GRP) ? a0 : 0.f;
                a.y = (n16 < GRP) ? a1 : 0.f;
            }
            const float* vr0 = (p0 == cur) ? vnew_row
                                           : (cache_v + kbase + (size_t)p0 * HEADD);
            const float* vr1 = (p1 == cur) ? vnew_row
                                           : (cache_v + kbase + (size_t)p1 * HEADD);
            v2f bb;             // B[k=p, n=d] (coalesced along d)
            bb.x = vr0[dbase + n16];
            bb.y = vr1[dbase + n16];
            acc = wmma_f32_k4(a, bb, acc);
        }
        if (lane < 16) {
            #pragma unroll
            for (int r = 0; r < GRP; ++r)
                attn_red[wave][r][dbase + lane] = acc[r];
        }
        __syncthreads();   // barrier after every nt; attn_red d-slices per nt are disjoint
    }
    __syncthreads();

    // cross-wave reduction + normalization
    for (int i = tid; i < GRP * HEADD; i += 256) {
        const int g = i >> 7, d = i & (HEADD - 1);
        float s = 0.f;
        #pragma unroll
        for (int w = 0; w < 8; ++w) s += attn_red[w][g][d];
        attn_buf[(size_t)b * (NH*HEADD) + (kv * GRP + g) * HEADD + d] = s * inv_l[g];
    }
}

// ---------------------------------------------------------------------------
// Kernel 3: output projection. out = attn(32x4096) @ wo^T (4096x4096).
// ---------------------------------------------------------------------------
__global__ __launch_bounds__(256)
void oproj_kernel(const float* __restrict__ attn_buf,
                  const float* __restrict__ wo,
                  float* __restrict__ out)
{
    const int wave = threadIdx.x >> 5;
    const int lane = threadIdx.x & 31;
    const int tile = blockIdx.x * 8 + wave;          // 0..511
    const int mbase = (tile & 1) * 16;
    const int nbase = (tile >> 1) * 16;
    const int n16   = lane & 15;
    const int c2    = (lane & 16) ? 2 : 0;
    const int col   = nbase + n16;

    const float* arow = attn_buf + (size_t)(mbase + n16) * HDIM + c2;
    const float* brow = wo + (size_t)col * HDIM + c2;

    v8f acc = {};
    #pragma unroll 8
    for (int k = 0; k < HDIM; k += 4) {
        v2f a = *(const v2f*)(arow + k);
        v2f b = *(const v2f*)(brow + k);
        acc = wmma_f32_k4(a, b, acc);
    }

    const int rowoff = (lane & 16) ? 8 : 0;
    #pragma unroll
    for (int r = 0; r < 8; ++r)
        out[(size_t)(mbase + r + rowoff) * HDIM + col] = acc[r];
}

// ---------------------------------------------------------------------------
extern "C" void kernel_launch(void* const* d_in, const int* in_sizes, int n_in,
                              void* d_out, int out_size, void* d_ws, size_t ws_size,
                              hipStream_t stream)
{
    const float* x         = (const float*)d_in[0];
    const float* wq        = (const float*)d_in[1];
    const float* wk        = (const float*)d_in[2];
    const float* wv        = (const float*)d_in[3];
    const float* wo        = (const float*)d_in[4];
    const float* rot       = (const float*)d_in[5];
    const float* cache_k   = (const float*)d_in[6];
    const float* cache_v   = (const float*)d_in[7];
    const float* attn_mask = (const float*)d_in[8];
    const int*   start_pos = (const int*)d_in[9];
    const int*   cur_pos   = (const int*)d_in[10];
    float* out = (float*)d_out;

    float* ws       = (float*)d_ws;
    float* q_buf    = ws;                         // 32*4096
    float* k_new    = ws + 131072;                // 32*1024
    float* v_new    = ws + 163840;                // 32*1024
    float* attn_buf = ws + 196608;                // 32*4096

    // 1) QKV + RoPE: 768 tiles / 8 waves -> 96 blocks
    qkv_rope_kernel<<<96, 256, 0, stream>>>(x, wq, wk, wv, rot,
                                            q_buf, k_new, v_new);
    // 2) attention: one block per (kv, b)
    attn_kernel<<<dim3(NKV, BATCH), 256, 0, stream>>>(
        cache_k, cache_v, attn_mask, start_pos, cur_pos,
        q_buf, k_new, v_new, attn_buf);
    // 3) output projection: 512 tiles / 8 waves -> 64 blocks
    oproj_kernel<<<64, 256, 0, stream>>>(attn_buf, wo, out);
}